// VoxelGrid_26491358282267
// MI455X (gfx1250) — compile-verified
//
#include <hip/hip_runtime.h>
#include <cstdint>
#include <cstddef>

typedef float v4f __attribute__((ext_vector_type(4)));

#define NV    100          // VOXEL_SIZE
#define NB    8            // BATCH
#define NPTS  65536        // NUM_COORDS (power of two -> shift for batch id)
#define OUTC  10           // output channels: 3 coord-mean, 3 feat-mean, 3 idx, 1 occ

// ---------------------------------------------------------------------------
// Kernel 1: zero the whole output buffer (in-place accumulators).
// 320 MB streaming write, b128 per lane, NT hint (output > 192 MB L2).
// ---------------------------------------------------------------------------
__global__ void __launch_bounds__(256) vg_zero(v4f* __restrict__ out, long long n4) {
  long long i = (long long)blockIdx.x * 256 + threadIdx.x;
  if (i < n4) {
    v4f z = (v4f)0.0f;
    __builtin_nontemporal_store(z, out + i);
  }
}

// ---------------------------------------------------------------------------
// Kernel 2: scatter points into the voxel accumulators.
// AoS (x,y,z)/(f0,f1,f2) point data staged through LDS with CDNA5 async
// global->LDS copies (coalesced dword streams, ASYNCcnt-tracked), then
// per-point gather from LDS and 7 f32 L2 atomics per point.
// ---------------------------------------------------------------------------
__global__ void __launch_bounds__(256) vg_scatter(const float* __restrict__ coords,
                                                  const float* __restrict__ feats,
                                                  float* __restrict__ out) {
  __shared__ float sC[256 * 3];
  __shared__ float sF[256 * 3];
  const int t = threadIdx.x;
  const long long base = (long long)blockIdx.x * 256;       // first point of tile
  const unsigned cbaseBytes = (unsigned)(base * 3 * 4);     // <= ~6.3 MB, fits u32

#pragma unroll
  for (int k = 0; k < 3; ++k) {
    unsigned goff = cbaseBytes + (unsigned)((k * 256 + t) * 4);
    unsigned ldsC = (unsigned)(size_t)(&sC[k * 256 + t]);   // low 32 bits = LDS byte addr
    unsigned ldsF = (unsigned)(size_t)(&sF[k * 256 + t]);
    asm volatile("global_load_async_to_lds_b32 %0, %1, %2"
                 :: "v"(ldsC), "v"(goff), "s"(coords) : "memory");
    asm volatile("global_load_async_to_lds_b32 %0, %1, %2"
                 :: "v"(ldsF), "v"(goff), "s"(feats) : "memory");
  }
  asm volatile("s_wait_asynccnt 0" ::: "memory");
  __syncthreads();

  // Bit-identical to the JAX f32 constant math:
  const float resv   = (1.0f - 0.0f) / (100.0f + 1e-12f);   // == 0.01f
  const float denom  = resv + 1e-12f;
  const float shiftv = 0.0f - resv;                         // bb_mins_shifted

  const float cx = sC[t * 3 + 0];
  const float cy = sC[t * 3 + 1];
  const float cz = sC[t * 3 + 2];
  int ix = (int)floorf((cx - shiftv) / denom);
  int iy = (int)floorf((cy - shiftv) / denom);
  int iz = (int)floorf((cz - shiftv) / denom);

  // Grid cells 0 and 101 are the discarded boundary shell -> skip them.
  if (ix >= 1 && ix <= NV && iy >= 1 && iy <= NV && iz >= 1 && iz <= NV) {
    const int b = (int)((base + t) >> 16);                  // NPTS == 2^16
    const long long o =
        ((((long long)b * NV + (ix - 1)) * NV + (iy - 1)) * NV + (iz - 1)) * OUTC;
    atomicAdd(out + o + 0, cx);
    atomicAdd(out + o + 1, cy);
    atomicAdd(out + o + 2, cz);
    atomicAdd(out + o + 3, sF[t * 3 + 0]);
    atomicAdd(out + o + 4, sF[t * 3 + 1]);
    atomicAdd(out + o + 5, sF[t * 3 + 2]);
    atomicAdd(out + o + 9, 1.0f);                           // count
  }
}

// ---------------------------------------------------------------------------
// Kernel 3: normalize sums -> means, fill index grid + occupancy.
// Tile of 256 voxels (2560 floats) staged through LDS so every global access
// is a unit-stride dword stream:
//   async global->LDS load, in-place per-voxel compute in LDS,
//   async LDS->global NT store (CDNA5 DMA path, no VGPR round-trip).
// ---------------------------------------------------------------------------
__global__ void __launch_bounds__(256) vg_finalize(float* __restrict__ out) {
  __shared__ float s[256 * OUTC];                 // 10 KB
  const int t = threadIdx.x;
  const int vbase = blockIdx.x * 256;             // first voxel of tile
  const unsigned gbaseBytes = (unsigned)vbase * (OUTC * 4u);  // < 2^31

#pragma unroll
  for (int k = 0; k < OUTC; ++k) {
    unsigned goff = gbaseBytes + (unsigned)((k * 256 + t) * 4);
    unsigned lds  = (unsigned)(size_t)(&s[k * 256 + t]);
    asm volatile("global_load_async_to_lds_b32 %0, %1, %2"
                 :: "v"(lds), "v"(goff), "s"(out) : "memory");
  }
  asm volatile("s_wait_asynccnt 0" ::: "memory");
  __syncthreads();

  // Thread t owns voxel vbase+t -> its private 10-float record in LDS.
  const int v = vbase + t;
  const int r = v % (NV * NV * NV);
  const int z = r % NV;
  const int y = (r / NV) % NV;
  const int x = r / (NV * NV);

  float* m = &s[t * OUTC];
  const float cnt = m[9];
  const float inv = 1.0f / fmaxf(cnt, 1.0f);
  m[0] *= inv;  m[1] *= inv;  m[2] *= inv;
  m[3] *= inv;  m[4] *= inv;  m[5] *= inv;
  m[6] = (float)x / 100.0f;                       // match x / V in f32
  m[7] = (float)y / 100.0f;
  m[8] = (float)z / 100.0f;
  m[9] = cnt > 0.0f ? 1.0f : 0.0f;                // occupancy
  __syncthreads();

#pragma unroll
  for (int k = 0; k < OUTC; ++k) {
    unsigned goff = gbaseBytes + (unsigned)((k * 256 + t) * 4);
    unsigned lds  = (unsigned)(size_t)(&s[k * 256 + t]);
    asm volatile("global_store_async_from_lds_b32 %0, %1, %2 th:TH_STORE_NT"
                 :: "v"(goff), "v"(lds), "s"(out) : "memory");
  }
  asm volatile("s_wait_asynccnt 0" ::: "memory"); // S_ENDPGM also drains, belt+braces
}

// ---------------------------------------------------------------------------
extern "C" void kernel_launch(void* const* d_in, const int* in_sizes, int n_in,
                              void* d_out, int out_size, void* d_ws, size_t ws_size,
                              hipStream_t stream) {
  (void)in_sizes; (void)n_in; (void)d_ws; (void)ws_size;
  const float* coords = (const float*)d_in[0];
  const float* feats  = (const float*)d_in[1];
  float* out = (float*)d_out;

  // out_size = 8 * 100^3 * 10 = 80,000,000 floats (divisible by 4)
  const long long n4 = (long long)out_size / 4;
  const int zeroBlocks = (int)((n4 + 255) / 256);
  vg_zero<<<zeroBlocks, 256, 0, stream>>>((v4f*)out, n4);

  const int scatterBlocks = (NB * NPTS) / 256;    // 2048
  vg_scatter<<<scatterBlocks, 256, 0, stream>>>(coords, feats, out);

  const int nvox = NB * NV * NV * NV;             // 8,000,000 (multiple of 256)
  const int finBlocks = nvox / 256;               // 31250
  vg_finalize<<<finBlocks, 256, 0, stream>>>(out);
}